// RNN_77000173683522
// MI455X (gfx1250) — compile-verified
//
#include <hip/hip_runtime.h>
#include <math.h>

#define DIMS 2048
#define SEQ  4096
#define NBLK 64          // persistent blocks for the recurrence (co-resident)
#define THREADS 256      // 8 waves of 32 on CDNA5

#define MT 128           // block tile rows (inputs rows)
#define NT 64            // block tile cols (W_hi rows)
#define KT 64            // K stage depth

typedef __attribute__((ext_vector_type(2))) float v2f;
typedef __attribute__((ext_vector_type(8))) float v8f;
typedef __attribute__((ext_vector_type(4))) unsigned int v4u;
typedef __attribute__((ext_vector_type(8))) int v8i;
typedef __attribute__((ext_vector_type(4))) int v4i;

// ---------------------------------------------------------------------------
// TDM tile load: DMA a 2D tile of `rows` x KT f32 (row stride = DIMS elements)
// from global memory into LDS at byte offset lds_off, row-major contiguous.
// D# packing per CDNA5 ISA ch.8 (group0 128b, group1 256b, groups 2/3 unused
// for 2D tiles -> zeros). Tracked with TENSORcnt.
// ---------------------------------------------------------------------------
#if __has_builtin(__builtin_amdgcn_tensor_load_to_lds)
#define HAVE_TDM 1
__device__ __forceinline__ void tdm_load_tile(unsigned lds_off, const float* gptr,
                                              int rows) {
    unsigned long long ga = (unsigned long long)(size_t)gptr;
    v4u g0;
    g0.x = 1u;                                          // count=1, user mode
    g0.y = lds_off;                                     // lds_addr (bytes)
    g0.z = (unsigned)(ga & 0xFFFFFFFFu);                // global_addr[31:0]
    g0.w = (unsigned)((ga >> 32) & 0x01FFFFFFu)         // global_addr[56:32]
         | (2u << 30);                                  // type=2 ("image")
    v8i g1;
    g1[0] = (int)(2u << 16);          // wg_mask=0 | data_size=2 (4 bytes)
    g1[1] = (int)(2048u << 16);       // atomic_bar_addr=0 | tensor_dim0[15:0]=2048
    g1[2] = (int)(4096u << 16);       // tensor_dim0[31:16]=0 | tensor_dim1[15:0]=4096
    g1[3] = (int)((unsigned)KT << 16);// tensor_dim1[31:16]=0 | tile_dim0=KT
    g1[4] = rows;                     // tile_dim1=rows | tile_dim2=0
    g1[5] = (int)DIMS;                // tensor_dim0_stride[31:0] = 2048
    g1[6] = 0;                        // stride hi | tensor_dim1_stride lo (unused, 2D)
    g1[7] = 0;
    v4i gz4 = {0, 0, 0, 0};           // groups 2/3: unused for 2D tiles
    v8i gz8 = {0, 0, 0, 0, 0, 0, 0, 0};
    __builtin_amdgcn_tensor_load_to_lds(g0, g1, gz4, gz4, gz8, 0);
}
#else
#define HAVE_TDM 0
#endif

// ---------------------------------------------------------------------------
// Phase 1: Z = inputs @ W_hi^T + b  (4096x2048x2048 f32, WMMA + TDM staging)
//
// Grid: (SEQ/MT) x (DIMS/NT) = 32 x 32 = 1024 blocks, 8 waves each.
// Per K stage, wave 0 DMAs the A tile (MT x KT of inputs) and the B tile
// (NT rows of W_hi x KT) into LDS via the Tensor Data Mover, double-buffered
// on TENSORcnt so stage k+1 streams while stage k feeds the matrix pipes.
// Wave w owns a 32x32 register tile (4 f32 accumulators): row band (w&3),
// column half (w>>2). Per 4-deep k step: 4 fragment loads -> 4 WMMAs (2x
// operand reuse in both M and N).
// ---------------------------------------------------------------------------
__global__ void __launch_bounds__(THREADS)
rnn_gemm_whi_wmma(const float* __restrict__ A,    // inputs [SEQ, DIMS]
                  const float* __restrict__ W,    // W_hi   [DIMS, DIMS]
                  const float* __restrict__ bias, // b      [DIMS]
                  float* __restrict__ Z)          // out    [SEQ, DIMS]
{
    __shared__ float sA[2][MT * KT];   // 2 x 32 KB
    __shared__ float sB[2][NT * KT];   // 2 x 16 KB

    const int waveId = threadIdx.x >> 5;
    const int lane   = threadIdx.x & 31;
    const int half   = lane >> 4;          // 0: K{0,1}, 1: K{2,3} of each step
    const int l16    = lane & 15;

    const int br = blockIdx.x >> 5;        // 0..31  row-tile  (inputs)
    const int bc = blockIdx.x & 31;        // 0..31  col-tile  (W_hi)

    const int mi = waveId & 3;             // 32-row band within block tile
    const int nj = waveId >> 2;            // which 32-col half

    const float* aG = A + (size_t)(br * MT) * DIMS;   // tile row base
    const float* bG = W + (size_t)(bc * NT) * DIMS;

#if HAVE_TDM
    const unsigned offA[2] = { (unsigned)(size_t)&sA[0][0], (unsigned)(size_t)&sA[1][0] };
    const unsigned offB[2] = { (unsigned)(size_t)&sB[0][0], (unsigned)(size_t)&sB[1][0] };
#endif

    v8f c00 = {}, c01 = {}, c10 = {}, c11 = {};
    const int NK = DIMS / KT;              // 32 K stages

#if HAVE_TDM
    if (waveId == 0) {                     // TDM is wave-level; issue once
        tdm_load_tile(offA[0], aG, MT);
        tdm_load_tile(offB[0], bG, NT);
    }
#endif

    for (int kt = 0; kt < NK; ++kt) {
        const int cur = kt & 1;
#if HAVE_TDM
        if (waveId == 0) {
            if (kt + 1 < NK) {             // prefetch next stage into other buffer
                tdm_load_tile(offA[cur ^ 1], aG + (kt + 1) * KT, MT);
                tdm_load_tile(offB[cur ^ 1], bG + (kt + 1) * KT, NT);
                __builtin_amdgcn_s_wait_tensorcnt(2);  // stage kt resident
            } else {
                __builtin_amdgcn_s_wait_tensorcnt(0);
            }
        }
#else
        // Fallback staging: cooperative 128-bit loads.
        for (int i = threadIdx.x; i < MT * KT / 4; i += THREADS) {
            const int r = i / (KT / 4), cq = i % (KT / 4);
            ((float4*)&sA[cur][0])[i] = *(const float4*)(aG + (size_t)r * DIMS + kt * KT + cq * 4);
        }
        for (int i = threadIdx.x; i < NT * KT / 4; i += THREADS) {
            const int r = i / (KT / 4), cq = i % (KT / 4);
            ((float4*)&sB[cur][0])[i] = *(const float4*)(bG + (size_t)r * DIMS + kt * KT + cq * 4);
        }
#endif
        __syncthreads();                   // publish LDS tiles to all waves

        const float* a_ld0 = &sA[cur][(mi * 32 + l16) * KT + (half << 1)];
        const float* a_ld1 = a_ld0 + 16 * KT;
        const float* b_ld0 = &sB[cur][(nj * 32 + l16) * KT + (half << 1)];
        const float* b_ld1 = b_ld0 + 16 * KT;

        #pragma unroll
        for (int kk = 0; kk < KT; kk += 4) {
            float2 a0f = *(const float2*)(a_ld0 + kk);
            float2 a1f = *(const float2*)(a_ld1 + kk);
            float2 b0f = *(const float2*)(b_ld0 + kk);
            float2 b1f = *(const float2*)(b_ld1 + kk);
            v2f a0; a0.x = a0f.x; a0.y = a0f.y;
            v2f a1; a1.x = a1f.x; a1.y = a1f.y;
            v2f b0; b0.x = b0f.x; b0.y = b0f.y;
            v2f b1; b1.x = b1f.x; b1.y = b1f.y;
            c00 = __builtin_amdgcn_wmma_f32_16x16x4_f32(false, a0, false, b0,
                                                        (short)0, c00, false, false);
            c01 = __builtin_amdgcn_wmma_f32_16x16x4_f32(false, a0, false, b1,
                                                        (short)0, c01, false, false);
            c10 = __builtin_amdgcn_wmma_f32_16x16x4_f32(false, a1, false, b0,
                                                        (short)0, c10, false, false);
            c11 = __builtin_amdgcn_wmma_f32_16x16x4_f32(false, a1, false, b1,
                                                        (short)0, c11, false, false);
        }
        __syncthreads();                   // all waves done before buffer reuse
    }

    // Epilogue: C/D layout lane 0-15 -> (M=v, N=l16), lanes 16-31 -> (M=v+8).
    const int col0 = bc * NT + nj * 32 + l16;
    const int col1 = col0 + 16;
    const int rowA = br * MT + mi * 32 + half * 8;      // c00 / c01 band
    const int rowB = rowA + 16;                         // c10 / c11 band
    const float bv0 = bias[col0];
    const float bv1 = bias[col1];
    float* zA0 = Z + (size_t)rowA * DIMS + col0;
    float* zA1 = Z + (size_t)rowA * DIMS + col1;
    float* zB0 = Z + (size_t)rowB * DIMS + col0;
    float* zB1 = Z + (size_t)rowB * DIMS + col1;
    #pragma unroll
    for (int v = 0; v < 8; ++v) {
        zA0[(size_t)v * DIMS] = c00[v] + bv0;
        zA1[(size_t)v * DIMS] = c01[v] + bv1;
        zB0[(size_t)v * DIMS] = c10[v] + bv0;
        zB1[(size_t)v * DIMS] = c11[v] + bv1;
    }
}

// ---------------------------------------------------------------------------
// Phase 2: serial recurrence. h_t = tanh(Z_t + W_hh h_{t-1}); out_t = x_t + h_t.
// Persistent NBLK blocks; block b owns output dims [b*32, b*32+32).
// 8 lanes cooperate per output dim over float4-strided K slices; h staged in
// LDS; W_hh streams from the 192 MB L2 (fully resident after step 0).
// Device-wide generation barrier with s_sleep backoff between steps.
// ---------------------------------------------------------------------------
__global__ void __launch_bounds__(THREADS)
rnn_recurrence(const float* __restrict__ x,
               const float* __restrict__ Whh,
               const float* __restrict__ Z,
               float* __restrict__ h,
               float* __restrict__ out,
               unsigned* __restrict__ bar)     // [2] {count, generation}
{
    __shared__ float hs[DIMS];
    const int tid = threadIdx.x;
    const int lk  = tid & 7;
    const int od  = tid >> 3;
    const int d   = blockIdx.x * 32 + od;
    const float* wrow = Whh + (size_t)d * DIMS;

    volatile unsigned* vgen = (volatile unsigned*)(bar + 1);

    for (int t = 0; t < SEQ; ++t) {
        for (int i = tid; i < DIMS / 4; i += THREADS) {
            ((float4*)hs)[i] = ((const float4*)h)[i];
        }
        __syncthreads();

        float acc = 0.f;
        #pragma unroll 4
        for (int j = 0; j < DIMS / 32; ++j) {
            const int k = (j << 5) + (lk << 2);
            const float4 w  = *(const float4*)(wrow + k);
            const float4 hv = *(const float4*)(hs + k);
            acc += w.x * hv.x + w.y * hv.y + w.z * hv.z + w.w * hv.w;
        }
        acc += __shfl_xor(acc, 1, 32);
        acc += __shfl_xor(acc, 2, 32);
        acc += __shfl_xor(acc, 4, 32);

        if (lk == 0) {
            const size_t idx = (size_t)t * DIMS + d;
            const float hnew = tanhf(Z[idx] + acc);
            h[d] = hnew;
            out[idx] = x[idx] + hnew;
        }

        __threadfence();
        __syncthreads();
        if (tid == 0) {
            unsigned arrived = atomicAdd(bar, 1u);
            if (arrived == NBLK - 1) {
                *bar = 0u;
                __threadfence();
                atomicAdd(bar + 1, 1u);
            } else {
                while (*vgen <= (unsigned)t) {
                    __builtin_amdgcn_s_sleep(1);
                }
            }
            __threadfence();
        }
        __syncthreads();
    }
}

// ---------------------------------------------------------------------------
// Launch: TDM/WMMA GEMM -> seed h with h0 -> zero barrier -> persistent scan.
// Workspace: Z (32 MB) | h (8 KB) | barrier counters.
// ---------------------------------------------------------------------------
extern "C" void kernel_launch(void* const* d_in, const int* in_sizes, int n_in,
                              void* d_out, int out_size, void* d_ws, size_t ws_size,
                              hipStream_t stream) {
    const float* inputs = (const float*)d_in[0];
    const float* W_hi   = (const float*)d_in[1];
    const float* W_hh   = (const float*)d_in[2];
    const float* bias   = (const float*)d_in[3];
    const float* h0     = (const float*)d_in[4];
    float* out          = (float*)d_out;

    char* ws = (char*)d_ws;
    float*    Z   = (float*)ws;
    float*    h   = (float*)(ws + (size_t)SEQ * DIMS * sizeof(float));
    unsigned* bar = (unsigned*)(ws + (size_t)SEQ * DIMS * sizeof(float)
                                   + (size_t)DIMS * sizeof(float));

    rnn_gemm_whi_wmma<<<(SEQ / MT) * (DIMS / NT), THREADS, 0, stream>>>(
        inputs, W_hi, bias, Z);

    (void)hipMemcpyAsync(h, h0, DIMS * sizeof(float), hipMemcpyDeviceToDevice, stream);
    (void)hipMemsetAsync(bar, 0, 2 * sizeof(unsigned), stream);

    rnn_recurrence<<<NBLK, THREADS, 0, stream>>>(inputs, W_hh, Z, h, out, bar);
}